// HashEncoder2D_57930518888962
// MI455X (gfx1250) — compile-verified
//
#include <hip/hip_runtime.h>
#include <hip/hip_bf16.h>
#include <cmath>
#include <cstdint>
#include <cstddef>

// ---------------- problem constants (from reference) ----------------
#define L_      16
#define T_      262144            // 2^18 table slots per level
#define TMASK_  (T_ - 1)
#define PI2_    2654435761u
#define NSTAGE  3                 // levels 0..2 staged in LDS (N=16,29,53 -> 4105 entries)
#define SMEM_ENTRIES 4224         // float2 entries; 4105 needed, padded
#define BLOCK_  512               // 16 waves/block; halves per-block TDM staging traffic

struct HEParams {
    int      NS[L_];              // per-level grid resolution
    unsigned ent_off[NSTAGE];     // LDS entry offset per staged level
    unsigned entries[NSTAGE];     // (N+1)^2 per staged level
    unsigned g0[NSTAGE][4];       // TDM D# group0 (lds_addr word patched in kernel)
    int      g1[NSTAGE][8];       // TDM D# group1
};

typedef unsigned int v4u_t __attribute__((ext_vector_type(4)));
typedef int          v8i_t __attribute__((ext_vector_type(8)));
typedef int          v4i_t __attribute__((ext_vector_type(4)));
typedef float        v2f_t __attribute__((ext_vector_type(2)));
typedef float        v4f_t __attribute__((ext_vector_type(4)));

__global__ __launch_bounds__(BLOCK_)
void hashenc2d_kernel(const float* __restrict__ xy,
                      const float* __restrict__ tables,
                      float*       __restrict__ out,
                      HEParams p, int nB)
{
    __shared__ float2 s_tab[SMEM_ENTRIES];

    // ---- stage dense low-res tables into LDS via the Tensor Data Mover ----
#if __has_builtin(__builtin_amdgcn_tensor_load_to_lds)
    if (threadIdx.x < 32u) {                       // wave 0 only (uniform branch)
        unsigned lbase = (unsigned)(uintptr_t)(void*)&s_tab[0]; // LDS byte offset
#pragma unroll
        for (int l = 0; l < NSTAGE; ++l) {
            v4u_t g0 = { p.g0[l][0], p.g0[l][1] + lbase, p.g0[l][2], p.g0[l][3] };
            v8i_t g1 = { p.g1[l][0], p.g1[l][1], p.g1[l][2], p.g1[l][3],
                         p.g1[l][4], p.g1[l][5], p.g1[l][6], p.g1[l][7] };
            v4i_t z4 = { 0, 0, 0, 0 };
#if __clang_major__ >= 23
            v8i_t z8 = { 0, 0, 0, 0, 0, 0, 0, 0 };
            __builtin_amdgcn_tensor_load_to_lds(g0, g1, z4, z4, z8, 0);
#else
            __builtin_amdgcn_tensor_load_to_lds(g0, g1, z4, z4, 0);
#endif
        }
        __builtin_amdgcn_s_wait_tensorcnt(0);      // TDM done before barrier signal
    }
#else
    // cooperative fallback if the TDM builtin is unavailable
    {
        const float2* gt = (const float2*)tables;
        for (int l = 0; l < NSTAGE; ++l) {
            unsigned n = p.entries[l], off = p.ent_off[l];
            const float2* src = gt + (size_t)l * T_;
            for (unsigned i = threadIdx.x; i < n; i += blockDim.x)
                s_tab[off + i] = src[i];
        }
    }
#endif
    __syncthreads();

    int b = (int)(blockIdx.x * blockDim.x + threadIdx.x);
    if (b >= nB) return;

    // xy is read exactly once across the whole dispatch -> non-temporal load
    v2f_t q = __builtin_nontemporal_load((const v2f_t*)xy + b);
    float x = fminf(fmaxf(q.x, 0.0f), 1.0f);
    float y = fminf(fmaxf(q.y, 0.0f), 1.0f);

    const float2* gt = (const float2*)tables;
    float o[2 * L_];

#pragma unroll
    for (int l = 0; l < L_; ++l) {
        int   Nl = p.NS[l];
        float fN = (float)Nl;
        float px = x * fN, py = y * fN;
        float fx = floorf(px), fy = floorf(py);
        int x0 = (int)fx, y0 = (int)fy;
        int x1 = min(x0 + 1, Nl), y1 = min(y0 + 1, Nl);
        float wx = px - fx, wy = py - fy;

        float2 f00, f10, f01, f11;
        if (l < NSTAGE) {                          // dense, LDS-resident (ds_load_b64)
            unsigned s   = (unsigned)Nl + 1u;
            unsigned off = p.ent_off[l];
            f00 = s_tab[off + (unsigned)x0 * s + (unsigned)y0];
            f10 = s_tab[off + (unsigned)x1 * s + (unsigned)y0];
            f01 = s_tab[off + (unsigned)x0 * s + (unsigned)y1];
            f11 = s_tab[off + (unsigned)x1 * s + (unsigned)y1];
        } else {
            const float2* tl = gt + (size_t)l * T_;
            unsigned i00, i10, i01, i11;
            if (Nl <= 511) {                       // (Nl+1)^2 <= T  -> dense
                unsigned s = (unsigned)Nl + 1u;
                i00 = (unsigned)x0 * s + (unsigned)y0;
                i10 = (unsigned)x1 * s + (unsigned)y0;
                i01 = (unsigned)x0 * s + (unsigned)y1;
                i11 = (unsigned)x1 * s + (unsigned)y1;
            } else {                               // spatial hash; low 18 bits match int64 ref
                unsigned hy0 = (unsigned)y0 * PI2_;
                unsigned hy1 = (unsigned)y1 * PI2_;
                i00 = ((unsigned)x0 ^ hy0) & TMASK_;
                i10 = ((unsigned)x1 ^ hy0) & TMASK_;
                i01 = ((unsigned)x0 ^ hy1) & TMASK_;
                i11 = ((unsigned)x1 ^ hy1) & TMASK_;
            }
            f00 = tl[i00]; f10 = tl[i10]; f01 = tl[i01]; f11 = tl[i11];
        }

        float iwx = 1.0f - wx, iwy = 1.0f - wy;
        float f0x = f00.x * iwx + f10.x * wx;
        float f0y = f00.y * iwx + f10.y * wx;
        float f1x = f01.x * iwx + f11.x * wx;
        float f1y = f01.y * iwx + f11.y * wx;
        o[2 * l + 0] = f0x * iwy + f1x * wy;
        o[2 * l + 1] = f0y * iwy + f1y * wy;
    }

    // 32 contiguous floats per point -> 8 x b128 NT stores (write-once stream:
    // keep the 256MB output out of L2 so the 32MB tables stay resident)
    v4f_t* o4 = (v4f_t*)(out + (size_t)b * (2 * L_));
#pragma unroll
    for (int k = 0; k < (2 * L_) / 4; ++k) {
        v4f_t v = { o[4 * k + 0], o[4 * k + 1], o[4 * k + 2], o[4 * k + 3] };
        __builtin_nontemporal_store(v, o4 + k);
    }
}

extern "C" void kernel_launch(void* const* d_in, const int* in_sizes, int n_in,
                              void* d_out, int out_size, void* d_ws, size_t ws_size,
                              hipStream_t stream)
{
    const float* xy     = (const float*)d_in[0];
    const float* tables = (const float*)d_in[1];
    float*       out    = (float*)d_out;
    int nB = in_sizes[0] / 2;

    HEParams p{};
    // identical double math to the Python reference (exp/log/pow/floor via libm)
    double bb = std::exp((std::log(131072.0) - std::log(16.0)) / 15.0);
    for (int l = 0; l < L_; ++l)
        p.NS[l] = (int)std::floor(16.0 * std::pow(bb, (double)l));

    unsigned off = 0;
    for (int l = 0; l < NSTAGE; ++l) {
        unsigned n1  = (unsigned)p.NS[l] + 1u;
        unsigned ent = n1 * n1;
        p.ent_off[l] = off;
        p.entries[l] = ent;

        uint64_t ga = (uint64_t)tables + (uint64_t)l * T_ * 2 * sizeof(float);
        // D# group0: count=1 | lds_addr | global_addr[56:0] | type=2
        p.g0[l][0] = 1u;
        p.g0[l][1] = off * 8u;                                  // + LDS base in kernel
        p.g0[l][2] = (unsigned)(ga & 0xFFFFFFFFu);
        p.g0[l][3] = (unsigned)((ga >> 32) & 0x1FFFFFFu) | (2u << 30);

        // D# group1: data_size=3 (8B), 2D tile: tile_dim0=ent, tile_dim1=1,
        // tensor_dim0=ent, tensor_dim1=1, stride0=ent (contiguous), no pad/multicast
        unsigned g1[8] = {0};
        g1[0] = 3u << 16;                        // data_size = 8 bytes
        g1[1] = (ent & 0xFFFFu) << 16;           // tensor_dim0[15:0]
        g1[2] = (ent >> 16) | (1u << 16);        // tensor_dim0[31:16] | tensor_dim1=1
        g1[3] = (ent & 0xFFFFu) << 16;           // tile_dim0 = ent (< 65536)
        g1[4] = 1u;                              // tile_dim1 = 1, tile_dim2 = 0
        g1[5] = ent;                             // tensor_dim0_stride[31:0]
        g1[6] = 0u;                              // stride0[47:32] | stride1[15:0]
        g1[7] = 0u;
        for (int k = 0; k < 8; ++k) p.g1[l][k] = (int)g1[k];

        off += ent;
    }

    dim3 block(BLOCK_);
    dim3 grid((unsigned)((nB + BLOCK_ - 1) / BLOCK_));
    hipLaunchKernelGGL(hashenc2d_kernel, grid, block, 0, stream,
                       xy, tables, out, p, nB);
}